// DCT2net_60662118089390
// MI455X (gfx1250) — compile-verified
//
#include <hip/hip_runtime.h>
#include <hip/hip_bf16.h>

// DCT2net fused pipeline for gfx1250 (MI455X), wave32 + WMMA f32 16x16x4.
//
// ws layout (floats), total ~4.06 MB:
//   xpad   : 4*280*280
//   Pm1pad : 176*176 (zero padded)
//   Qinvpad: 176*176 (zero padded, = inv(Pm1))
//   num    : 4*280*280
//   div    : 4*280*280
//   aug    : 169*344 (Gauss-Jordan scratch)

#define P_    13
#define PP    169
#define PPD   176
#define NIMG  4
#define HW    256
#define HP    280
#define HO    268
#define LPOS  (HO*HO)            /* 71824 */
#define IMGPAD (HP*HP)           /* 78400 */

#define XPAD_OFF 0
#define PM1_OFF  (NIMG*IMGPAD)                 /* 313600 */
#define QIN_OFF  (PM1_OFF + PPD*PPD)           /* 344576 */
#define NUM_OFF  (QIN_OFF + PPD*PPD)           /* 375552 */
#define DIV_OFF  (NUM_OFF + NIMG*IMGPAD)       /* 689152 */
#define AUG_OFF  (DIV_OFF + NIMG*IMGPAD)       /* 1002752 */
#define AUGW 344

#define WAVES 4

typedef float v2f __attribute__((ext_vector_type(2)));
typedef float v8f __attribute__((ext_vector_type(8)));

// ---------------------------------------------------------------------------
// 1) reflect-pad + scale, zero accumulators, zero-pad Pm1 to 176x176
// ---------------------------------------------------------------------------
__global__ void prep_kernel(const float* __restrict__ x,
                            const float* __restrict__ Pm1,
                            float* __restrict__ ws) {
    int idx = blockIdx.x * blockDim.x + threadIdx.x;
    if (idx < NIMG * IMGPAD) {
        ws[NUM_OFF + idx] = 0.0f;
        ws[DIV_OFF + idx] = 0.0f;
        int n   = idx / IMGPAD;
        int rem = idx % IMGPAD;
        int y   = rem / HP;
        int xx  = rem % HP;
        int ty = y  - (P_ - 1); ty = (ty < 0) ? -ty : ((ty > HW - 1) ? 2 * (HW - 1) - ty : ty);
        int tx = xx - (P_ - 1); tx = (tx < 0) ? -tx : ((tx > HW - 1) ? 2 * (HW - 1) - tx : tx);
        ws[XPAD_OFF + idx] = 2.0f * x[n * HW * HW + ty * HW + tx] - 1.0f;
    }
    if (idx < PPD * PPD) {
        int r = idx / PPD, c = idx % PPD;
        ws[PM1_OFF + idx] = (r < PP && c < PP) ? Pm1[r * PP + c] : 0.0f;
        ws[QIN_OFF + idx] = 0.0f;
    }
}

// ---------------------------------------------------------------------------
// 2) inv(Pm1) via Gauss-Jordan, single workgroup (169^3 ~ 4.8 MFLOP)
// ---------------------------------------------------------------------------
__global__ void invert_kernel(const float* __restrict__ Pm1,
                              float* __restrict__ ws) {
    float* aug = ws + AUG_OFF;
    const int tid = threadIdx.x;

    for (int idx = tid; idx < PP * AUGW; idx += blockDim.x) {
        int r = idx / AUGW, c = idx % AUGW;
        float v = 0.0f;
        if (c < PP)            v = Pm1[r * PP + c];
        else if (c < 2 * PP)   v = (c - PP == r) ? 1.0f : 0.0f;
        aug[idx] = v;
    }
    __syncthreads();

    for (int k = 0; k < PP; ++k) {
        float piv = aug[k * AUGW + k];
        __syncthreads();
        float inv = 1.0f / piv;
        for (int c = tid; c < 2 * PP; c += blockDim.x)
            aug[k * AUGW + c] *= inv;
        __syncthreads();
        for (int r = tid; r < PP; r += blockDim.x) {
            if (r != k) {
                float f = aug[r * AUGW + k];
                for (int c = 0; c < 2 * PP; ++c)
                    aug[r * AUGW + c] -= f * aug[k * AUGW + c];
            }
        }
        __syncthreads();
    }

    for (int idx = tid; idx < PP * PP; idx += blockDim.x) {
        int r = idx / PP, c = idx % PP;
        ws[QIN_OFF + r * PPD + c] = aug[r * AUGW + PP + c];
    }
}

// ---------------------------------------------------------------------------
// 3) fused:  t = Pm1 @ patches ; hard-threshold ; w = 1/(1+nnz) ;
//            rec = inv(Pm1) @ t ; atomically fold w*rec and w into num/div.
//    One wave = 16 patch columns; 2 x (44 K-steps x 11 M-tiles) f32 WMMAs.
//    B loads are branchless: zero-padded A columns (k>=169) annihilate any
//    finite garbage in B, and tail columns are discarded at the fold stage,
//    so no EXEC masking is ever needed around the WMMAs.
// ---------------------------------------------------------------------------
__global__ __launch_bounds__(32 * WAVES)
void dct2net_main(const float* __restrict__ sigma_,
                  float* __restrict__ ws) {
    __shared__ float lds_t[WAVES][PPD][17];   // 47,872 B

    const float* __restrict__ xpad = ws + XPAD_OFF;
    const float* __restrict__ pm1  = ws + PM1_OFF;
    const float* __restrict__ qin  = ws + QIN_OFF;
    float* __restrict__ numb = ws + NUM_OFF;
    float* __restrict__ divb = ws + DIV_OFF;

    const int wave  = threadIdx.x >> 5;
    const int lane  = threadIdx.x & 31;
    const int laneN = lane & 15;
    const int hi    = lane >> 4;         // 0: lanes 0-15, 1: lanes 16-31

    const int img  = blockIdx.y;
    const int l0   = (blockIdx.x * WAVES + wave) * 16;
    const int lcol = l0 + laneN;
    const bool colOK = (lcol < LPOS);
    const int lc = colOK ? lcol : 0;     // clamp tail columns to a valid patch
    const int ip = lc / HO;
    const int jp = lc - ip * HO;
    const float* __restrict__ xb = xpad + img * IMGPAD + ip * HP + jp;

    const float lam = 6.0f * sigma_[0];  // 3 * (2*sigma_)

    const v8f vzero = {0.f, 0.f, 0.f, 0.f, 0.f, 0.f, 0.f, 0.f};
    v8f acc[11];
#pragma unroll
    for (int mt = 0; mt < 11; ++mt) acc[mt] = vzero;

    // ---- GEMM 1: t = Pm1pad(176x176) @ patches(176x16) ----
#pragma unroll 4
    for (int kt = 0; kt < 44; ++kt) {
        const int ka = kt * 4 + 2 * hi;      // even -> 8B aligned A loads
        // Branchless B fragment: k in [169,176) reads a few floats past xpad
        // (still inside ws, finite) and is annihilated by zero A columns.
        const int d0 = ka / P_;
        const int d1 = (ka + 1) / P_;
        v2f b;
        b.x = xb[d0 * HP + (ka - d0 * P_)];
        b.y = xb[d1 * HP + (ka + 1 - d1 * P_)];
#pragma unroll
        for (int mt = 0; mt < 11; ++mt) {
            v2f a = *(const v2f*)(pm1 + (mt * 16 + laneN) * PPD + ka);
            acc[mt] = __builtin_amdgcn_wmma_f32_16x16x4_f32(
                false, a, false, b, (short)0, acc[mt], false, false);
        }
    }

    // ---- hard threshold + nonzero count (rows 1..168) ----
    int cnt = 0;
#pragma unroll
    for (int mt = 0; mt < 11; ++mt) {
#pragma unroll
        for (int v = 0; v < 8; ++v) {
            float t = acc[mt][v];
            t = (fabsf(t) > lam) ? t : 0.0f;
            acc[mt][v] = t;
            bool isRow0 = (mt == 0 && v == 0 && hi == 0);   // global row M==0
            cnt += (!isRow0 && t != 0.0f) ? 1 : 0;
        }
    }
    cnt += __shfl_xor(cnt, 16, 32);          // combine lanes N and N+16 (wave32)
    const float w = 1.0f / (1.0f + (float)cnt);

    // ---- re-lay t tile through LDS (C-layout -> B-fragment layout) ----
#pragma unroll
    for (int mt = 0; mt < 11; ++mt)
#pragma unroll
        for (int v = 0; v < 8; ++v)
            lds_t[wave][mt * 16 + v + hi * 8][laneN] = acc[mt][v];
    __syncthreads();

    // ---- GEMM 2: rec = Qinvpad(176x176) @ t(176x16) ----
    v8f acc2[11];
#pragma unroll
    for (int mt = 0; mt < 11; ++mt) acc2[mt] = vzero;

#pragma unroll 4
    for (int kt = 0; kt < 44; ++kt) {
        const int ka = kt * 4 + 2 * hi;
        v2f b;
        b.x = lds_t[wave][ka][laneN];
        b.y = lds_t[wave][ka + 1][laneN];
#pragma unroll
        for (int mt = 0; mt < 11; ++mt) {
            v2f a = *(const v2f*)(qin + (mt * 16 + laneN) * PPD + ka);
            acc2[mt] = __builtin_amdgcn_wmma_f32_16x16x4_f32(
                false, a, false, b, (short)0, acc2[mt], false, false);
        }
    }

    // ---- weighted overlap-add fold into num/div (L2-resident atomics) ----
    if (colOK) {
        float* nb = numb + img * IMGPAD + ip * HP + jp;
        float* db = divb + img * IMGPAD + ip * HP + jp;
#pragma unroll
        for (int mt = 0; mt < 11; ++mt) {
#pragma unroll
            for (int v = 0; v < 8; ++v) {
                int M = mt * 16 + v + hi * 8;
                if (M < PP) {
                    int d   = M / P_;
                    int off = d * HP + (M - d * P_);
                    atomicAdd(nb + off, acc2[mt][v] * w);
                    atomicAdd(db + off, w);
                }
            }
        }
    }
}

// ---------------------------------------------------------------------------
// 4) crop + normalize
// ---------------------------------------------------------------------------
__global__ void finish_kernel(const float* __restrict__ ws,
                              float* __restrict__ out) {
    int idx = blockIdx.x * blockDim.x + threadIdx.x;
    if (idx >= NIMG * HW * HW) return;
    int n   = idx / (HW * HW);
    int rem = idx % (HW * HW);
    int y   = rem / HW;
    int x   = rem % HW;
    int s = n * IMGPAD + (y + P_ - 1) * HP + (x + P_ - 1);
    out[idx] = (ws[NUM_OFF + s] / ws[DIV_OFF + s] + 1.0f) * 0.5f;
}

extern "C" void kernel_launch(void* const* d_in, const int* in_sizes, int n_in,
                              void* d_out, int out_size, void* d_ws, size_t ws_size,
                              hipStream_t stream) {
    const float* x      = (const float*)d_in[0];   // (4,1,256,256) f32
    const float* sigma_ = (const float*)d_in[1];   // (1,) f32
    const float* Pm1    = (const float*)d_in[2];   // (169,169) f32
    float* out = (float*)d_out;                    // (4,1,256,256) f32
    float* ws  = (float*)d_ws;                     // needs ~4.1 MB

    const int prepN = NIMG * IMGPAD;
    prep_kernel<<<(prepN + 255) / 256, 256, 0, stream>>>(x, Pm1, ws);

    invert_kernel<<<1, 256, 0, stream>>>(Pm1, ws);

    dim3 grid((LPOS + WAVES * 16 - 1) / (WAVES * 16), NIMG);
    dct2net_main<<<grid, 32 * WAVES, 0, stream>>>(sigma_, ws);

    const int outN = NIMG * HW * HW;
    finish_kernel<<<(outN + 255) / 256, 256, 0, stream>>>(ws, out);
}